// ASTSFormer_43636867728099
// MI455X (gfx1250) — compile-verified
//
#include <hip/hip_runtime.h>

#define S_TOT 4096     // superpixels per batch (64x64 tiles)
#define NW    64
#define PLANE 1048576  // H*W

typedef float v2f __attribute__((ext_vector_type(2)));
typedef float v8f __attribute__((ext_vector_type(8)));

__device__ __forceinline__ int offj(int j) { return ((j / 3) - 1) * NW + (j % 3) - 1; }

// ds_swizzle_b32 immediate SWAPX16: offset[14:10]=xor=0x10, offset[9:5]=or=0, offset[4:0]=and=0x1F
__device__ __forceinline__ float swap16(float x)
{
  return __int_as_float(__builtin_amdgcn_ds_swizzle(__float_as_int(x), 0x401F));
}

// Fill per-tile candidate tables in LDS:
//  sAm2[c*16+m] = -2*spf[c][cand_m]   (m<9, else 0)
//  sn[m]        = ||spf[:,cand_m]||^2 (m<9, else 0)
__device__ __forceinline__ void setup_cands(int t, int s, const float* __restrict__ spfb,
                                            float* sAm2, float* sn, int* scand, int* svalid)
{
  if (t < 16) {
    int m = t;
    float a0 = 0.f, a1 = 0.f, a2 = 0.f;
    if (m < 9) {
      int cand = s + offj(m);
      int vl = (cand >= 0) && (cand < S_TOT);
      int cd = cand < 0 ? 0 : (cand > S_TOT - 1 ? S_TOT - 1 : cand);
      a0 = spfb[cd];
      a1 = spfb[S_TOT + cd];
      a2 = spfb[2 * S_TOT + cd];
      scand[m] = cd;
      svalid[m] = vl;
    }
    sAm2[m]      = -2.f * a0;
    sAm2[16 + m] = -2.f * a1;
    sAm2[32 + m] = -2.f * a2;
    sn[m] = a0 * a0 + a1 * a1 + a2 * a2;
  }
}

// Per-lane squared distances of this lane's pixel to the 9 candidates.
// V_WMMA_F32_16X16X4_F32 with A rows duplicated (row m = cand m&7):
//   D[m][n] = sn[m&7] + sum_c (-2*spf[c][m&7]) * px[c][n]
// so each lane-half reads candidates 0..7 of its own pixels directly from its
// D VGPRs (no cross-lane moves). Candidate 8 is 3 scalar FMAs. Only the three
// channel half-swaps need a lane exchange (immediate ds_swizzle SWAPX16).
__device__ __forceinline__ void dist9(const float* sAm2, const float* sn,
                                      float ch0, float ch1, float ch2,
                                      int lane, float dist[9])
{
  const int hi = lane >> 4;
  const int m8 = lane & 7;                 // duplicated candidate row index

  v2f a;
  a.x = sAm2[(hi ? 32 : 0) + m8];          // low half: chan0, high half: chan2
  a.y = hi ? 0.f : sAm2[16 + m8];          // low half: chan1, high half: K-pad

  v8f c;
#pragma unroll
  for (int r = 0; r < 8; ++r) c[r] = sn[r];   // C[m][n] = sn[m&7]

  float sx0 = swap16(ch0);
  float sx1 = swap16(ch1);
  float sx2 = swap16(ch2);

  v2f b0, b1;                              // B: K across lane-halves/vgprs
  b0.x = hi ? sx2 : ch0;  b0.y = hi ? 0.f : ch1;   // pixels = lanes 0..15
  b1.x = hi ? ch2 : sx0;  b1.y = hi ? 0.f : sx1;   // pixels = lanes 16..31

  v8f d0 = __builtin_amdgcn_wmma_f32_16x16x4_f32(false, a, false, b0, (short)0, c, false, false);
  v8f d1 = __builtin_amdgcn_wmma_f32_16x16x4_f32(false, a, false, b1, (short)0, c, false, false);

  float pn = ch0 * ch0 + ch1 * ch1 + ch2 * ch2;
#pragma unroll
  for (int j = 0; j < 8; ++j)
    dist[j] = (hi ? d1[j] : d0[j]) + pn;   // own half's pixels, cands 0..7

  // candidate 8: scalar dot (sAm2 already holds -2*spf)
  float d8 = sn[8];
  d8 = fmaf(sAm2[8],  ch0, d8);
  d8 = fmaf(sAm2[24], ch1, d8);
  d8 = fmaf(sAm2[40], ch2, d8);
  dist[8] = d8 + pn;
}

// Kernel A: 16x16 block means -> cent (spf layout [b][c][s])
__global__ void cent_kernel(const float* __restrict__ px, float* __restrict__ spf)
{
  int blk = blockIdx.x;
  int b = blk >> 12, s = blk & (S_TOT - 1);
  int ti = s >> 6, tj = s & 63;
  int t = threadIdx.x;
  __shared__ float acc[3];
  if (t < 3) acc[t] = 0.f;
  __syncthreads();
  int n = ((ti << 4) + (t >> 4)) * 1024 + (tj << 4) + (t & 15);
  const float* pxb = px + (size_t)b * 3 * PLANE;
  atomicAdd(&acc[0], pxb[n]);
  atomicAdd(&acc[1], pxb[PLANE + n]);
  atomicAdd(&acc[2], pxb[2 * PLANE + n]);
  __syncthreads();
  if (t < 3) spf[((size_t)b * 3 + t) * S_TOT + s] = acc[t] * (1.f / 256.f);
}

// Kernel B: one EM iteration — softmax affinities, segment sums into num/den
__global__ void iter_kernel(const float* __restrict__ px, const float* __restrict__ spf,
                            float* __restrict__ num, float* __restrict__ den)
{
  int blk = blockIdx.x;
  int b = blk >> 12, s = blk & (S_TOT - 1);
  int ti = s >> 6, tj = s & 63;
  int t = threadIdx.x;

  __shared__ float sAm2[48], sn[16];
  __shared__ int scand[9], svalid[9];
  __shared__ float binN[27], binD[9];

  setup_cands(t, s, spf + (size_t)b * 3 * S_TOT, sAm2, sn, scand, svalid);
  if (t >= 64 && t < 100) {
    int u = t - 64;
    if (u < 27) binN[u] = 0.f; else binD[u - 27] = 0.f;
  }
  __syncthreads();

  int lane = t & 31;
  int n = ((ti << 4) + (t >> 4)) * 1024 + (tj << 4) + (t & 15);
  const float* pxb = px + (size_t)b * 3 * PLANE;
  float ch0 = pxb[n], ch1 = pxb[PLANE + n], ch2 = pxb[2 * PLANE + n];

  float dist[9];
  dist9(sAm2, sn, ch0, ch1, ch2, lane, dist);

  float dmin = 3.0e38f;
#pragma unroll
  for (int j = 0; j < 9; ++j) {
    dist[j] = svalid[j] ? dist[j] : 1.0e16f;
    dmin = fminf(dmin, dist[j]);
  }
  float e[9], sum = 0.f;
#pragma unroll
  for (int j = 0; j < 9; ++j) {
    e[j] = svalid[j] ? __expf(dmin - dist[j]) : 0.f;
    sum += e[j];
  }
  float inv = 1.f / sum;
#pragma unroll
  for (int j = 0; j < 9; ++j) {
    float w = e[j] * inv;
    atomicAdd(&binN[j * 3 + 0], w * ch0);
    atomicAdd(&binN[j * 3 + 1], w * ch1);
    atomicAdd(&binN[j * 3 + 2], w * ch2);
    atomicAdd(&binD[j], w);
  }
  __syncthreads();
  if (t < 36) {
    int j = t >> 2, k = t & 3;
    size_t base = (size_t)b * S_TOT + scand[j];
    if (k < 3) atomicAdd(&num[base * 3 + k], binN[j * 3 + k]);
    else       atomicAdd(&den[base], binD[j]);
  }
}

// Kernel C: spf2[b][c][s] = num[b][s][c] / (den[b][s] + 1e-16)
__global__ void spf_update_kernel(const float* __restrict__ num, const float* __restrict__ den,
                                  float* __restrict__ spf2)
{
  int idx = blockIdx.x * 256 + threadIdx.x;
  if (idx < 2 * S_TOT) {
    int b = idx >> 12, s = idx & (S_TOT - 1);
    float d = 1.f / (den[idx] + 1e-16f);
    for (int c2 = 0; c2 < 3; ++c2)
      spf2[((size_t)b * 3 + c2) * S_TOT + s] = num[(size_t)idx * 3 + c2] * d;
  }
}

// Kernel D: batch-0 hard assignment + per-label sums/counts
__global__ void assign_kernel(const float* __restrict__ px, const float* __restrict__ spf2,
                              float* __restrict__ sums, float* __restrict__ cnts)
{
  int s = blockIdx.x;
  int ti = s >> 6, tj = s & 63;
  int t = threadIdx.x;

  __shared__ float sAm2[48], sn[16];
  __shared__ int scand[9], svalid[9];
  __shared__ float binS[27], binC[9];

  setup_cands(t, s, spf2, sAm2, sn, scand, svalid);
  if (t >= 64 && t < 100) {
    int u = t - 64;
    if (u < 27) binS[u] = 0.f; else binC[u - 27] = 0.f;
  }
  __syncthreads();

  int lane = t & 31;
  int n = ((ti << 4) + (t >> 4)) * 1024 + (tj << 4) + (t & 15);
  float ch0 = px[n], ch1 = px[PLANE + n], ch2 = px[2 * PLANE + n];

  float dist[9];
  dist9(sAm2, sn, ch0, ch1, ch2, lane, dist);

  float dmin = 3.0e38f; int jb = 0;
#pragma unroll
  for (int j = 0; j < 9; ++j) {
    float dj = svalid[j] ? dist[j] : 1.0e16f;
    if (dj < dmin) { dmin = dj; jb = j; }
  }
  atomicAdd(&binS[jb * 3 + 0], ch0);
  atomicAdd(&binS[jb * 3 + 1], ch1);
  atomicAdd(&binS[jb * 3 + 2], ch2);
  atomicAdd(&binC[jb], 1.f);
  __syncthreads();
  if (t < 36) {
    int j = t >> 2, k = t & 3;
    int cand = scand[j];
    if (k < 3) atomicAdd(&sums[(size_t)cand * 3 + k], binS[j * 3 + k]);
    else       atomicAdd(&cnts[cand], binC[j]);
  }
}

// Kernel E: means = sums / max(cnts, 1)
__global__ void means_kernel(const float* __restrict__ sums, const float* __restrict__ cnts,
                             float* __restrict__ means)
{
  int s = blockIdx.x * 256 + threadIdx.x;
  if (s < S_TOT) {
    float inv = 1.f / fmaxf(cnts[s], 1.f);
    for (int c2 = 0; c2 < 3; ++c2)
      means[(size_t)s * 3 + c2] = sums[(size_t)s * 3 + c2] * inv;
  }
}

// Kernel F: recompute labels, gather means -> output image (1,3,1024,1024)
__global__ void gather_kernel(const float* __restrict__ px, const float* __restrict__ spf2,
                              const float* __restrict__ means, float* __restrict__ out)
{
  int s = blockIdx.x;
  int ti = s >> 6, tj = s & 63;
  int t = threadIdx.x;

  __shared__ float sAm2[48], sn[16];
  __shared__ int scand[9], svalid[9];
  __shared__ float mLds[27];

  setup_cands(t, s, spf2, sAm2, sn, scand, svalid);
  if (t >= 64 && t < 91) {
    int u = t - 64;
    int j = u / 3, c2 = u % 3;
    int cand = s + offj(j);
    cand = cand < 0 ? 0 : (cand > S_TOT - 1 ? S_TOT - 1 : cand);
    mLds[u] = means[(size_t)cand * 3 + c2];
  }
  __syncthreads();

  int lane = t & 31;
  int n = ((ti << 4) + (t >> 4)) * 1024 + (tj << 4) + (t & 15);
  float ch0 = px[n], ch1 = px[PLANE + n], ch2 = px[2 * PLANE + n];

  float dist[9];
  dist9(sAm2, sn, ch0, ch1, ch2, lane, dist);

  float dmin = 3.0e38f; int jb = 0;
#pragma unroll
  for (int j = 0; j < 9; ++j) {
    float dj = svalid[j] ? dist[j] : 1.0e16f;
    if (dj < dmin) { dmin = dj; jb = j; }
  }
  out[n]             = mLds[jb * 3 + 0];
  out[PLANE + n]     = mLds[jb * 3 + 1];
  out[2 * PLANE + n] = mLds[jb * 3 + 2];
}

extern "C" void kernel_launch(void* const* d_in, const int* in_sizes, int n_in,
                              void* d_out, int out_size, void* d_ws, size_t ws_size,
                              hipStream_t stream)
{
  const float* px = (const float*)d_in[0];
  float* out = (float*)d_out;
  float* ws = (float*)d_ws;

  // workspace layout (floats)
  float* spf   = ws;             // 24576  [b][c][s] initial centroids
  float* spf2  = ws + 24576;     // 24576  [b][c][s] updated centroids
  float* means = ws + 49152;     // 12288  [s][c]
  float* num   = ws + 61440;     // 24576  [b][s][c]  (accumulator)
  float* den   = ws + 86016;     // 8192   [b][s]     (accumulator)
  float* sums  = ws + 94208;     // 12288  [s][c]     (accumulator)
  float* cnts  = ws + 106496;    // 4096   [s]        (accumulator)

  // zero all accumulators (contiguous range) — capture-safe async memset
  hipMemsetAsync(num, 0, (size_t)49152 * sizeof(float), stream);

  cent_kernel<<<dim3(2 * S_TOT), dim3(256), 0, stream>>>(px, spf);
  iter_kernel<<<dim3(2 * S_TOT), dim3(256), 0, stream>>>(px, spf, num, den);
  spf_update_kernel<<<dim3(32), dim3(256), 0, stream>>>(num, den, spf2);
  assign_kernel<<<dim3(S_TOT), dim3(256), 0, stream>>>(px, spf2, sums, cnts);
  means_kernel<<<dim3(16), dim3(256), 0, stream>>>(sums, cnts, means);
  gather_kernel<<<dim3(S_TOT), dim3(256), 0, stream>>>(px, spf2, means, out);
}